// TokenGNN_2800318677072
// MI455X (gfx1250) — compile-verified
//
#include <hip/hip_runtime.h>

typedef __attribute__((ext_vector_type(2))) float v2f;
typedef __attribute__((ext_vector_type(8))) float v8f;

#define DIM 128

// ----------------- degree / normalization -----------------
__global__ void k_init_deg(float* deg, int n) {
    int i = blockIdx.x * blockDim.x + threadIdx.x;
    if (i < n) deg[i] = 1.0f;  // self-loop contributes 1
}

__global__ void k_count_deg(const int* __restrict__ col, float* deg, int e) {
    int i = blockIdx.x * blockDim.x + threadIdx.x;
    if (i < e) atomicAdd(&deg[col[i]], 1.0f);
}

__global__ void k_dinv(float* deg, int n) {
    int i = blockIdx.x * blockDim.x + threadIdx.x;
    if (i < n) deg[i] = rsqrtf(deg[i]);
}

// ----------------- fp32 WMMA GEMM: Out = (relu?)X @ W + b -----------------
// One wave computes a 16-row x 128-col strip.
// W staged in LDS PRE-PAIRED along K: sWp[kk*128 + col] = {W[2kk][col], W[2kk+1][col]}
// so each B fragment is a single aligned ds_load_b64 directly into the WMMA
// operand pair (no v_mov shuffling).
template <int RELU>
__global__ __launch_bounds__(256, 2) void k_gemm(
    const float* __restrict__ X, const float* __restrict__ W,
    const float* __restrict__ bias, float* __restrict__ Out, int nRows) {
    __shared__ float2 sWp[DIM * DIM / 2];
    // cooperative staged+swizzled load of W (64KB)
    for (int p = threadIdx.x; p < DIM * DIM / 2; p += blockDim.x) {
        int kk  = p >> 7;        // K-pair index 0..63
        int col = p & 127;
        float2 w2;
        w2.x = W[(2 * kk)     * DIM + col];
        w2.y = W[(2 * kk + 1) * DIM + col];
        sWp[p] = w2;
    }
    __syncthreads();

    int wave = threadIdx.x >> 5;
    int lane = threadIdx.x & 31;
    int tile = blockIdx.x * 8 + wave;        // 16-row tile index
    if (tile * 16 >= nRows) return;          // nRows % 16 == 0 here
    int row0 = tile * 16;
    int l15  = lane & 15;
    int half = lane >> 4;                    // 0 or 1

    // 8 accumulators (16x16 each) covering N=0..127; init with bias (per-column)
    v8f acc[8];
#pragma unroll
    for (int t = 0; t < 8; ++t) {
        float bv = bias[t * 16 + l15];
#pragma unroll
        for (int v = 0; v < 8; ++v) acc[t][v] = bv;
    }

    // A fragment: lane l15 holds row (row0+l15); half=0 -> K,K+1, half=1 -> K+2,K+3
    const float* xrow = X + (size_t)(row0 + l15) * DIM;

#pragma unroll 4
    for (int k = 0; k < DIM; k += 4) {
        float2 a2 = *(const float2*)(xrow + k + 2 * half);
        v2f a;
        a[0] = a2.x;
        a[1] = a2.y;
        if (RELU) { a[0] = fmaxf(a[0], 0.0f); a[1] = fmaxf(a[1], 0.0f); }
        int kk = (k >> 1) + half;            // K-pair row for this half-wave
#pragma unroll
        for (int t = 0; t < 8; ++t) {
            float2 b2 = sWp[kk * DIM + t * 16 + l15];  // single ds_load_b64
            v2f b;
            b[0] = b2.x;
            b[1] = b2.y;
            acc[t] = __builtin_amdgcn_wmma_f32_16x16x4_f32(
                false, a, false, b, (short)0, acc[t], false, false);
        }
    }

    // C/D layout: VGPR v -> (M=v, N=lane) lanes 0-15 ; (M=v+8, N=lane-16) lanes 16-31
#pragma unroll
    for (int t = 0; t < 8; ++t) {
#pragma unroll
        for (int v = 0; v < 8; ++v) {
            int r = row0 + v + 8 * half;
            Out[(size_t)r * DIM + t * 16 + l15] = acc[t][v];
        }
    }
}

// ----------------- aggregation -----------------
// Init with self-loop term: Agg[i] = dinv[i]^2 * H[i]   (one float4 per thread)
__global__ void k_selfloop(const float* __restrict__ H, const float* __restrict__ dinv,
                           float* __restrict__ Agg, int n) {
    int idx = blockIdx.x * blockDim.x + threadIdx.x;   // over n * 32 float4s
    if (idx >= n * (DIM / 4)) return;
    int node = idx >> 5;
    float s = dinv[node];
    s = s * s;
    float4 v = ((const float4*)H)[idx];
    v.x *= s; v.y *= s; v.z *= s; v.w *= s;
    ((float4*)Agg)[idx] = v;
}

// One wave per edge: lane loads float4 of H[row], scales, atomically adds to Agg[col].
__global__ void k_scatter(const float* __restrict__ H, const int* __restrict__ row,
                          const int* __restrict__ col, const float* __restrict__ dinv,
                          float* __restrict__ Agg, int nE) {
    int e = blockIdx.x * (blockDim.x >> 5) + (threadIdx.x >> 5);
    if (e >= nE) return;
    int lane = threadIdx.x & 31;
    int r = row[e], c = col[e];
    float nrm = dinv[r] * dinv[c];
    float4 v = ((const float4*)(H + (size_t)r * DIM))[lane];
    float* dst = Agg + (size_t)c * DIM + lane * 4;
    atomicAdd(dst + 0, v.x * nrm);
    atomicAdd(dst + 1, v.y * nrm);
    atomicAdd(dst + 2, v.z * nrm);
    atomicAdd(dst + 3, v.w * nrm);
}

// ----------------- final ReLU + column mean -----------------
__global__ void k_zero(float* out) { if (threadIdx.x < DIM) out[threadIdx.x] = 0.0f; }

__global__ void k_mean(const float* __restrict__ Agg, float* __restrict__ out, int n) {
    __shared__ float red[256];
    int t = threadIdx.x;
    int c  = t & (DIM - 1);
    int rh = t >> 7;                       // 0 or 1: row interleave
    int chunk = (n + gridDim.x - 1) / gridDim.x;
    int start = blockIdx.x * chunk;
    int end = min(n, start + chunk);
    float s = 0.0f;
    for (int i = start + rh; i < end; i += 2)
        s += fmaxf(Agg[(size_t)i * DIM + c], 0.0f);
    red[t] = s;
    __syncthreads();
    if (t < DIM) atomicAdd(&out[c], (red[t] + red[t + DIM]) * (1.0f / (float)n));
}

// ----------------- launch -----------------
extern "C" void kernel_launch(void* const* d_in, const int* in_sizes, int n_in,
                              void* d_out, int out_size, void* d_ws, size_t ws_size,
                              hipStream_t stream) {
    const float* x  = (const float*)d_in[0];
    const int*   ei = (const int*)d_in[1];
    const float* W1 = (const float*)d_in[2];
    const float* b1 = (const float*)d_in[3];
    const float* W2 = (const float*)d_in[4];
    const float* b2 = (const float*)d_in[5];
    float* out = (float*)d_out;

    int n = in_sizes[0] / DIM;       // 100000
    int e = in_sizes[1] / 2;         // 1600000
    const int* rowp = ei;            // edge_index[0] = sources
    const int* colp = ei + e;        // edge_index[1] = targets

    size_t NH = (size_t)n * DIM;
    float* bufA = (float*)d_ws;      // h (post-GEMM)
    float* bufB = bufA + NH;         // aggregated
    float* dinv = bufB + NH;         // deg -> dinv in place

    int tB = 256;
    // 1) degree + rsqrt
    k_init_deg<<<(n + tB - 1) / tB, tB, 0, stream>>>(dinv, n);
    k_count_deg<<<(e + tB - 1) / tB, tB, 0, stream>>>(colp, dinv, e);
    k_dinv<<<(n + tB - 1) / tB, tB, 0, stream>>>(dinv, n);

    int gemmBlocks = (((n + 15) / 16) + 7) / 8;          // 8 waves/block, 1 tile/wave
    int slBlocks = (n * (DIM / 4) + tB - 1) / tB;
    int scBlocks = (e + 7) / 8;                          // 8 edges per 256-thread block

    // Layer 1: h1 = x@W1+b1 ; agg1 = D^-1/2 (A+I) D^-1/2 h1
    k_gemm<0><<<gemmBlocks, tB, 0, stream>>>(x, W1, b1, bufA, n);
    k_selfloop<<<slBlocks, tB, 0, stream>>>(bufA, dinv, bufB, n);
    k_scatter<<<scBlocks, tB, 0, stream>>>(bufA, rowp, colp, dinv, bufB, e);

    // Layer 2: h2 = relu(agg1)@W2+b2 (ReLU fused into A-load) ; agg2 likewise
    k_gemm<1><<<gemmBlocks, tB, 0, stream>>>(bufB, W2, b2, bufA, n);
    k_selfloop<<<slBlocks, tB, 0, stream>>>(bufA, dinv, bufB, n);
    k_scatter<<<scBlocks, tB, 0, stream>>>(bufA, rowp, colp, dinv, bufB, e);

    // Final: out = mean over nodes of relu(agg2)
    k_zero<<<1, 128, 0, stream>>>(out);
    k_mean<<<256, tB, 0, stream>>>(bufB, out, n);
}